// GCN_58394375356710
// MI455X (gfx1250) — compile-verified
//
#include <hip/hip_runtime.h>

typedef __attribute__((ext_vector_type(2))) float v2f;
typedef __attribute__((ext_vector_type(8))) float v8f;

#define N_NODES   100000
#define N_EDGES   3200000
#define D_FEAT    512
#define HIDDEN    16
#define N_CLASSES 40
#define NT2       3            // ceil(40/16) column tiles for layer-2 GEMM
#define M_TILES   (N_NODES/16) // 6250, exact

// ---------------------------------------------------------------- utilities
__global__ void k_fill(float* __restrict__ p, float v, int n) {
  int i = blockIdx.x * blockDim.x + threadIdx.x;
  if (i < n) p[i] = v;
}

// deg[dst] += 1 per edge (deg pre-filled with 1.0 for the self loop)
__global__ void k_deg(const int* __restrict__ e, float* __restrict__ deg) {
  int i = blockIdx.x * blockDim.x + threadIdx.x;
  if (i < N_EDGES) atomicAdd(&deg[e[N_EDGES + i]], 1.0f);
}

__global__ void k_rsqrt_inplace(float* __restrict__ d, int n) {
  int i = blockIdx.x * blockDim.x + threadIdx.x;
  if (i < n) {
    float v = d[i];
    d[i] = (v > 0.0f) ? rsqrtf(v) : 0.0f;
  }
}

// ------------------------------------------- layer 1 GEMM: h_lin = x @ W1
// One wave computes a 16x16 tile: 16 node rows x 16 hidden cols, K=512 in
// steps of 4 via v_wmma_f32_16x16x4_f32 (128 WMMAs per wave).
__global__ void k_gemm1(const float* __restrict__ x, const float* __restrict__ W,
                        float* __restrict__ out) {
  int lane = threadIdx.x & 31;
  int wave = threadIdx.x >> 5;
  int tile = blockIdx.x * (blockDim.x >> 5) + wave;
  if (tile >= M_TILES) return;          // wave-uniform: EXEC stays all-ones

  int lc = lane & 15;                   // A: row index (M); B: col index (N)
  int kh = lane >> 4;                   // which K-pair this half-wave holds

  const float* xrow = x + (size_t)(tile * 16 + lc) * D_FEAT;
  v8f acc = {};
  for (int k = 0; k < D_FEAT; k += 4) {
    v2f a = *(const v2f*)(xrow + k + 2 * kh);              // A[M=lc][K..K+1]
    v2f b;
    b.x = W[(k + 2 * kh    ) * HIDDEN + lc];               // B[K][N=lc]
    b.y = W[(k + 2 * kh + 1) * HIDDEN + lc];
    acc = __builtin_amdgcn_wmma_f32_16x16x4_f32(
        /*neg_a=*/false, a, /*neg_b=*/false, b,
        /*c_mod=*/(short)0, acc, /*reuse_a=*/false, /*reuse_b=*/false);
  }
  // C/D layout: VGPR v -> M = v + 8*kh, N = lc
  float* orow = out + (size_t)(tile * 16 + kh * 8) * HIDDEN + lc;
#pragma unroll
  for (int v = 0; v < 8; ++v) orow[(size_t)v * HIDDEN] = acc[v];
}

// ------------------------------- layer 1 scatter: h_agg[dst] += norm*h_lin[src]
__global__ void k_agg1(const int* __restrict__ e, const float* __restrict__ dinv,
                       const float* __restrict__ hl, float* __restrict__ hagg) {
  int i = blockIdx.x * blockDim.x + threadIdx.x;
  if (i >= N_EDGES) return;
  int s = e[i], d = e[N_EDGES + i];
  float nrm = dinv[s] * dinv[d];
  const float4* hs = (const float4*)(hl + (size_t)s * HIDDEN);
  float* hd = hagg + (size_t)d * HIDDEN;
#pragma unroll
  for (int q = 0; q < HIDDEN / 4; ++q) {
    float4 v = hs[q];
    atomicAdd(hd + q * 4 + 0, v.x * nrm);
    atomicAdd(hd + q * 4 + 1, v.y * nrm);
    atomicAdd(hd + q * 4 + 2, v.z * nrm);
    atomicAdd(hd + q * 4 + 3, v.w * nrm);
  }
}

// self loop + bias + ReLU (in place on h_agg)
__global__ void k_post1(const float* __restrict__ hl, const float* __restrict__ dinv,
                        const float* __restrict__ b1, float* __restrict__ hagg) {
  int i = blockIdx.x * blockDim.x + threadIdx.x;
  if (i >= N_NODES * HIDDEN) return;
  int node = i >> 4, f = i & 15;
  float di = dinv[node];
  float v = hagg[i] + hl[i] * di * di + b1[f];
  hagg[i] = v > 0.0f ? v : 0.0f;
}

// ------------------------------------------- layer 2 GEMM: h2 = h @ W2 (cols padded to 48)
__global__ void k_gemm2(const float* __restrict__ h, const float* __restrict__ W,
                        float* __restrict__ out) {
  int lane = threadIdx.x & 31;
  int wave = threadIdx.x >> 5;
  int t = blockIdx.x * (blockDim.x >> 5) + wave;
  if (t >= M_TILES * NT2) return;       // wave-uniform
  int tm = t / NT2;
  int tn = t - tm * NT2;

  int lc = lane & 15;
  int kh = lane >> 4;
  int colg = tn * 16 + lc;

  const float* hrow = h + (size_t)(tm * 16 + lc) * HIDDEN;
  v8f acc = {};
#pragma unroll
  for (int k = 0; k < HIDDEN; k += 4) {
    v2f a = *(const v2f*)(hrow + k + 2 * kh);
    float b0 = 0.0f, b1v = 0.0f;
    if (colg < N_CLASSES) {             // pad cols 40..47 with zeros
      b0  = W[(k + 2 * kh    ) * N_CLASSES + colg];
      b1v = W[(k + 2 * kh + 1) * N_CLASSES + colg];
    }
    v2f b; b.x = b0; b.y = b1v;         // control flow reconverged before WMMA
    acc = __builtin_amdgcn_wmma_f32_16x16x4_f32(
        false, a, false, b, (short)0, acc, false, false);
  }
  if (colg < N_CLASSES) {
    float* orow = out + (size_t)(tm * 16 + kh * 8) * N_CLASSES + colg;
#pragma unroll
    for (int v = 0; v < 8; ++v) orow[(size_t)v * N_CLASSES] = acc[v];
  }
}

// ------------------------------- layer 2 scatter into d_out
__global__ void k_agg2(const int* __restrict__ e, const float* __restrict__ dinv,
                       const float* __restrict__ h2, float* __restrict__ out) {
  int i = blockIdx.x * blockDim.x + threadIdx.x;
  if (i >= N_EDGES) return;
  int s = e[i], d = e[N_EDGES + i];
  float nrm = dinv[s] * dinv[d];
  const float4* hs = (const float4*)(h2 + (size_t)s * N_CLASSES);
  float* od = out + (size_t)d * N_CLASSES;
#pragma unroll
  for (int q = 0; q < N_CLASSES / 4; ++q) {
    float4 v = hs[q];
    atomicAdd(od + q * 4 + 0, v.x * nrm);
    atomicAdd(od + q * 4 + 1, v.y * nrm);
    atomicAdd(od + q * 4 + 2, v.z * nrm);
    atomicAdd(od + q * 4 + 3, v.w * nrm);
  }
}

__global__ void k_post2(const float* __restrict__ h2, const float* __restrict__ dinv,
                        const float* __restrict__ b2, float* __restrict__ out) {
  int i = blockIdx.x * blockDim.x + threadIdx.x;
  if (i >= N_NODES * N_CLASSES) return;
  int node = i / N_CLASSES, f = i - node * N_CLASSES;
  float di = dinv[node];
  out[i] += h2[i] * di * di + b2[f];
}

// ----------------------------------------------------------------- launcher
extern "C" void kernel_launch(void* const* d_in, const int* in_sizes, int n_in,
                              void* d_out, int out_size, void* d_ws, size_t ws_size,
                              hipStream_t stream) {
  const float* x  = (const float*)d_in[0];
  const int*   ei = (const int*)d_in[1];      // [2, N_EDGES] row-major
  const float* W1 = (const float*)d_in[2];
  const float* b1 = (const float*)d_in[3];
  const float* W2 = (const float*)d_in[4];
  const float* b2 = (const float*)d_in[5];
  float* out = (float*)d_out;

  float* ws    = (float*)d_ws;
  float* dinv  = ws;                                    // N_NODES (deg -> dinv in place)
  float* h_lin = dinv + N_NODES;                        // N_NODES*HIDDEN
  float* h_agg = h_lin + (size_t)N_NODES * HIDDEN;      // N_NODES*HIDDEN (reused as ReLU'd h)
  float* h2    = h_agg + (size_t)N_NODES * HIDDEN;      // N_NODES*N_CLASSES

  const int T = 256;
  // degrees: deg = 1 (self loop) + incoming count; then dinv = rsqrt(deg)
  k_fill<<<(N_NODES + T - 1) / T, T, 0, stream>>>(dinv, 1.0f, N_NODES);
  k_deg<<<(N_EDGES + T - 1) / T, T, 0, stream>>>(ei, dinv);
  k_rsqrt_inplace<<<(N_NODES + T - 1) / T, T, 0, stream>>>(dinv, N_NODES);

  // layer 1
  k_gemm1<<<(M_TILES + 3) / 4, 128, 0, stream>>>(x, W1, h_lin);
  k_fill<<<(N_NODES * HIDDEN + T - 1) / T, T, 0, stream>>>(h_agg, 0.0f, N_NODES * HIDDEN);
  k_agg1<<<(N_EDGES + T - 1) / T, T, 0, stream>>>(ei, dinv, h_lin, h_agg);
  k_post1<<<(N_NODES * HIDDEN + T - 1) / T, T, 0, stream>>>(h_lin, dinv, b1, h_agg);

  // layer 2
  k_gemm2<<<(M_TILES * NT2 + 3) / 4, 128, 0, stream>>>(h_agg, W2, h2);
  k_fill<<<(N_NODES * N_CLASSES + T - 1) / T, T, 0, stream>>>(out, 0.0f, N_NODES * N_CLASSES);
  k_agg2<<<(N_EDGES + T - 1) / T, T, 0, stream>>>(ei, dinv, h2, out);
  k_post2<<<(N_NODES * N_CLASSES + T - 1) / T, T, 0, stream>>>(h2, dinv, b2, out);
}